// BoundaryRenderingModule_32341103739184
// MI455X (gfx1250) — compile-verified
//
#include <hip/hip_runtime.h>
#include <hip/hip_bf16.h>
#include <stdint.h>

// Problem constants (fixed by the reference): x is (B=2, C=4, D=64, H=256, W=256) f32.
#define BB   2
#define CC   4
#define DD   64
#define HH   256
#define WW   256
#define HW   (HH * WW)          // 65536
#define DHW  (DD * HH * WW)     // 4194304  (= 1 << 22)

// ---------------- CDNA5 feature detection (device pass only) ----------------
#if defined(__HIP_DEVICE_COMPILE__) && __has_builtin(__builtin_amdgcn_global_load_async_to_lds_b64)
#define USE_ASYNC 1
#endif

#if defined(USE_ASYNC)
#define AS1 __attribute__((address_space(1)))
#define AS3 __attribute__((address_space(3)))
typedef int v2i __attribute__((vector_size(2 * sizeof(int))));  // matches builtin signature
#if __has_builtin(__builtin_amdgcn_s_wait_asynccnt)
#define WAIT_ASYNC(n) __builtin_amdgcn_s_wait_asynccnt(n)
#else
#define WAIT_ASYNC(n) asm volatile("s_wait_asynccnt %0" ::"i"(n) : "memory")
#endif
#else
#define WAIT_ASYNC(n) ((void)0)
#endif

#if defined(__HIP_DEVICE_COMPILE__) && __has_builtin(__builtin_amdgcn_ballot_w32)
#define BALLOT32(p) __builtin_amdgcn_ballot_w32(p)
#else
#define BALLOT32(p) ((unsigned)__ballot(p))
#endif

// Order-preserving float <-> uint mapping so uint atomicMin/Max give float min/max.
__device__ __forceinline__ unsigned ordKey(float f) {
  unsigned u = __float_as_uint(f);
  return (u & 0x80000000u) ? ~u : (u | 0x80000000u);
}
__device__ __forceinline__ float ordInv(unsigned o) {
  unsigned u = (o & 0x80000000u) ? (o ^ 0x80000000u) : ~o;
  return __uint_as_float(u);
}

// ---------------- Kernel 0: init workspace ----------------
// ws[0..3] = per-channel min key (init 0xFFFFFFFF), ws[4..7] = max key (init 0).
__global__ void brm_init_ws(unsigned* ws) {
  int t = threadIdx.x;
  if (t < 8) ws[t] = (t < 4) ? 0xFFFFFFFFu : 0u;
}

// ---------------- Kernel 1: per-channel min/max ----------------
// grid: (512, 4) blocks of 256 threads; float4 grid-stride over the channel.
// Regular (RT) loads on purpose: x is 128 MB vs 192 MB L2 -> this pass warms L2
// for the stencil pass.
__global__ void brm_minmax(const float* __restrict__ x, unsigned* __restrict__ ws) {
  const int c   = blockIdx.y;
  const int tid = threadIdx.x;
  float mn = 3.4028235e38f, mx = -3.4028235e38f;

  const unsigned nf4    = (2u * DHW) / 4u;            // float4 per channel (2 batches)
  const unsigned stride = gridDim.x * blockDim.x;
  for (unsigned i = blockIdx.x * blockDim.x + tid; i < nf4; i += stride) {
    unsigned b   = i >> 20;                           // DHW/4 = 1<<20 float4 per (b,c) chunk
    unsigned rem = i & ((1u << 20) - 1u);
    const float4* p = (const float4*)(x + ((size_t)(b * CC + c) << 22)) + rem;
    // Prefetch this thread's next grid-stride element (speculative; OOB dropped).
    unsigned ni = i + stride;
    const float4* np = (const float4*)(x + ((size_t)((ni >> 20) * CC + c) << 22)) + (ni & ((1u << 20) - 1u));
    __builtin_prefetch((const void*)np, 0, 0);        // global_prefetch_b8
    float4 v = *p;
    mn = fminf(mn, fminf(fminf(v.x, v.y), fminf(v.z, v.w)));
    mx = fmaxf(mx, fmaxf(fmaxf(v.x, v.y), fmaxf(v.z, v.w)));
  }
  // wave32 reduce
  for (int off = 16; off > 0; off >>= 1) {
    mn = fminf(mn, __shfl_xor(mn, off, 32));
    mx = fmaxf(mx, __shfl_xor(mx, off, 32));
  }
  __shared__ float smn[8], smx[8];
  const int wv = tid >> 5, ln = tid & 31;
  if (ln == 0) { smn[wv] = mn; smx[wv] = mx; }
  __syncthreads();
  if (tid == 0) {
    for (int i = 1; i < 8; ++i) { mn = fminf(mn, smn[i]); mx = fmaxf(mx, smx[i]); }
    atomicMin(&ws[c], ordKey(mn));
    atomicMax(&ws[4 + c], ordKey(mx));
  }
}

// ---------------- Kernel 2: fused binarize + 3x3x3 dilation + boundary ----------------
// Block = 256 threads (8 waves). Tile: DT x HT x 256 (full W). Rolls along D with a
// 3-slice ring of bit-packed masks; input planes staged via async global->LDS DMA
// (double buffered, 9 async instructions per wave per plane -> s_wait_asynccnt 9).
// Output stores are non-temporal so the 128 MB of writes don't evict the L2-resident
// input (input 128 MB + output 128 MB > 192 MB L2).
#define HT 16
#define DT 16
#define SROWS (HT + 2)   // 18 rows (H halo)
#define SLOT(p) (((p) + 192) % 3)

__global__ void brm_boundary(const float* __restrict__ x,
                             const unsigned* __restrict__ ws,
                             float* __restrict__ out) {
  __shared__ float    stage[2][SROWS * WW];   // 2 * 18 * 256 * 4B = 36864 B
  __shared__ unsigned ring[3][SROWS * 8];     // 3 * 18 * 8 * 4B  =  1728 B

  const int tid  = threadIdx.x;
  const int wv   = tid >> 5;
  const int lane = tid & 31;
  const int h0   = blockIdx.x * HT;
  const int d0   = blockIdx.y * DT;
  const int bc   = blockIdx.z;                // b*4 + c
  const int c    = bc & 3;
  const size_t chanBase = (size_t)bc * (size_t)DHW;

  const float mnv = ordInv(ws[c]);
  const float mxv = ordInv(ws[4 + c]);
  const float thr = mnv + 0.5f * (mxv - mnv); // binary = (x > thr)

  // Stage one H-halo'd plane (18 rows x 256 f32 = 18432 B) of depth-plane p into stage[buf].
  // 2304 8-byte chunks -> 9 async instructions per wave (uniform -> clean ASYNCcnt math).
  auto issue_load = [&](int p, int buf) {
    const int pc = min(max(p, 0), DD - 1);    // clamp (OOB rows are mask-zeroed later)
    const float* gb = x + chanBase + (size_t)pc * HW;
    for (int it = 0; it < 9; ++it) {
      const int id  = it * 256 + tid;         // chunk id, 0..2303
      const int row = id >> 7;                // 128 chunks per 256-float row
      const int col = (id & 127) << 1;        // even float offset -> 8B aligned
      const int h   = h0 - 1 + row;
      const int hc  = min(max(h, 0), HH - 1);
      const float* gp = gb + hc * WW + col;
      float*       lp = &stage[buf][row * WW + col];
#if defined(USE_ASYNC)
      __builtin_amdgcn_global_load_async_to_lds_b64(
          (AS1 v2i*)(AS1 void*)gp, (AS3 v2i*)(AS3 void*)lp, 0, 0);
#else
      lp[0] = gp[0];
      lp[1] = gp[1];
#endif
    }
  };

  // Threshold + bit-pack plane p (stage[buf]) into ring[SLOT(p)] via wave32 ballot.
  auto build = [&](int p, int buf) {
    unsigned* dst = ring[SLOT(p)];
    const bool dok = (p >= 0) && (p < DD);
    for (int i = 0; i < 18; ++i) {            // 144 words / 8 waves
      const int wordId = wv * 18 + i;
      const int row = wordId >> 3;
      const int ww  = wordId & 7;
      const int h   = h0 - 1 + row;
      const float v = stage[buf][row * WW + (ww << 5) + lane];
      unsigned m = BALLOT32(v > thr);
      if (!(dok && h >= 0 && h < HH)) m = 0u; // SAME padding == zeros outside
      if (lane == 0) dst[row * 8 + ww] = m;
    }
  };

  // Boundary of center plane q from ring slices q-1,q,q+1; coalesced NT f32 stores.
  auto compute = [&](int q) {
    const unsigned* rs[3] = { ring[SLOT(q - 1)], ring[SLOT(q)], ring[SLOT(q + 1)] };
    float* ob = out + chanBase + (size_t)q * HW;
    for (int i = 0; i < 16; ++i) {            // 128 words / 8 waves
      const int wordId = wv * 16 + i;
      const int rc = wordId >> 3;             // 0..HT-1
      const int ww = wordId & 7;
      const int r  = rc + 1;                  // slice row (halo offset)
      unsigned dil = 0u;
      #pragma unroll
      for (int s = 0; s < 3; ++s) {
        const unsigned* base = rs[s];
        #pragma unroll
        for (int dr = -1; dr <= 1; ++dr) {
          const unsigned* row = base + (r + dr) * 8;
          const unsigned m   = row[ww];
          const unsigned lft = (ww > 0) ? row[ww - 1] : 0u;
          const unsigned rgt = (ww < 7) ? row[ww + 1] : 0u;
          dil |= m | (m << 1) | (m >> 1) | (lft >> 31) | (rgt << 31);
        }
      }
      const unsigned ctr = rs[1][r * 8 + ww];
      const unsigned bw  = dil & ~ctr;        // boundary bits
      const float val = (float)((bw >> lane) & 1u);
      __builtin_nontemporal_store(val, &ob[(h0 + rc) * WW + (ww << 5) + lane]);
    }
  };

  // Software pipeline: double-buffered async stage, 3-slice mask ring.
  issue_load(d0 - 1, 0);
  issue_load(d0,     1);
  for (int p = d0 - 1; p <= d0 + DT; ++p) {
    const int buf = (p - d0 + 1) & 1;
    if (p < d0 + DT) { WAIT_ASYNC(9); } else { WAIT_ASYNC(0); }  // plane p landed
    __syncthreads();
    build(p, buf);
    __syncthreads();
    if (p + 2 <= d0 + DT) issue_load(p + 2, buf);  // reuse buf after build(p)
    if (p >= d0 + 1) compute(p - 1);
  }
}

extern "C" void kernel_launch(void* const* d_in, const int* in_sizes, int n_in,
                              void* d_out, int out_size, void* d_ws, size_t ws_size,
                              hipStream_t stream) {
  const float* x  = (const float*)d_in[0];
  float*       out = (float*)d_out;
  unsigned*    ws  = (unsigned*)d_ws;   // 8 uints: [min keys c=0..3][max keys c=0..3]

  brm_init_ws<<<1, 32, 0, stream>>>(ws);
  brm_minmax<<<dim3(512, 4), 256, 0, stream>>>(x, ws);
  brm_boundary<<<dim3(HH / HT, DD / DT, BB * CC), 256, 0, stream>>>(x, ws, out);
}